// QSVT_19258633355992
// MI455X (gfx1250) — compile-verified
//
#include <hip/hip_runtime.h>

// ---------------------------------------------------------------------------
// QSVT Chebyshev kernel for MI455X (gfx1250, wave32).
//
// Closed form derived from the reference circuit:
//   out[b, 4*(d-1)+0] = cos(x0)
//   out[b, 4*(d-1)+1] = cos(x1) * ( ps0 + pc0 * sum_j G_j * (2*T_d(x_j)^2 - 1) )
//   out[b, 4*(d-1)+2] = cos(x2)
//   out[b, 4*(d-1)+3] = cos(x3)
// with pc_i = (1+cos x_i)/2, ps_i = (1-cos x_i)/2,
//      G = {pc2*pc3, pc2*ps3, ps2*pc3, ps2*ps3}, T_d = Chebyshev.
//
// Memory-bound streaming kernel (~21 MB total @ 23.3 TB/s ~ 0.9 us).
// Data movement: CDNA5 Tensor Data Mover stages each workgroup's 4 KB x-tile
// into LDS (tensor_load_to_lds + s_wait_tensorcnt), read back with an explicit
// ds_load_b128 (inline asm so the compiler cannot fold the load of memory it
// never saw written); output streams out via non-temporal b128 stores.
// cos on [0,1) via even Taylor poly to x^10 (|err| <= 1/12! ~ 2.1e-9).
// ---------------------------------------------------------------------------

typedef float        v4f   __attribute__((ext_vector_type(4)));
typedef unsigned int u32x4 __attribute__((ext_vector_type(4)));
typedef int          i32x4 __attribute__((ext_vector_type(4)));
typedef int          i32x8 __attribute__((ext_vector_type(8)));

#if defined(__HIP_DEVICE_COMPILE__) && defined(__gfx1250__) && \
    __has_builtin(__builtin_amdgcn_tensor_load_to_lds) &&      \
    __has_builtin(__builtin_amdgcn_s_wait_tensorcnt)
#define QSVT_TDM 1
#else
#define QSVT_TDM 0
#endif

// cos(x) for x in [0,1): even Taylor series through x^10.
// Truncation error <= 1/12! ~ 2.1e-9, below fp32 roundoff for O(1) results.
static __device__ __forceinline__ float cos01(float x) {
  const float u = x * x;
  float p = fmaf(u, -2.7557319e-07f, 2.4801587e-05f);  // -1/10!, 1/8!
  p = fmaf(u, p, -1.3888889e-03f);                     // -1/6!
  p = fmaf(u, p, 4.1666668e-02f);                      //  1/4!
  p = fmaf(u, p, -5.0000000e-01f);                     // -1/2!
  return fmaf(u, p, 1.0f);
}

__global__ __launch_bounds__(256) void qsvt_cheb_kernel(
    const float* __restrict__ x, float* __restrict__ out, int n) {
  const int tid = threadIdx.x;
  const int e   = (int)blockIdx.x * 256 + tid;

  v4f xv;

#if QSVT_TDM
  __shared__ __align__(16) float xs[1024];  // 256 elements * 4 floats = 4 KB
  const bool full = ((int)blockIdx.x * 256 + 256) <= n;
  if (full) {
    if (tid < 32) {  // wave 0 issues the TDM DMA once per workgroup
      unsigned long long ga =
          (unsigned long long)(size_t)x + (unsigned long long)blockIdx.x * 4096ull;
      unsigned ldso = (unsigned)(size_t)(const void*)xs;  // LDS byte offset

      // D# group 0: count=1 | lds_addr | global_addr[56:0] | type=2
      u32x4 g0;
      g0[0] = 1u;                                            // count = 1 (valid)
      g0[1] = ldso;                                          // lds_addr
      g0[2] = (unsigned)(ga & 0xFFFFFFFFull);                // global_addr[31:0]
      g0[3] = (unsigned)((ga >> 32) & 0x01FFFFFFull)         // global_addr[56:32]
              | 0x80000000u;                                 // type = 2 ("image")

      // D# group 1: data_size=4B, tensor_dim0=1024, tensor_dim1=1,
      //             tile_dim0=1024, tile_dim1=1, tensor_dim0_stride=1024
      i32x8 g1;
      g1[0] = 0x00020000;          // workgroup_mask=0, data_size=2 (4 bytes)
      g1[1] = (int)(1024u << 16);  // tensor_dim0[15:0]=1024 (atomic_bar_addr=0)
      g1[2] = (int)(1u << 16);     // tensor_dim0[31:16]=0, tensor_dim1[15:0]=1
      g1[3] = (int)(1024u << 16);  // tensor_dim1[31:16]=0, tile_dim0=1024
      g1[4] = 1;                   // tile_dim1=1, tile_dim2=0
      g1[5] = 1024;                // tensor_dim0_stride[31:0]
      g1[6] = 0;                   // stride0[47:32]=0, stride1 low=0
      g1[7] = 0;

      i32x4 z4 = {0, 0, 0, 0};
#if defined(__clang_major__) && (__clang_major__ >= 23)
      i32x8 z8 = {0, 0, 0, 0, 0, 0, 0, 0};
      __builtin_amdgcn_tensor_load_to_lds(g0, g1, z4, z4, z8, 0);
#else
      __builtin_amdgcn_tensor_load_to_lds(g0, g1, z4, z4, 0);
#endif
      __builtin_amdgcn_s_wait_tensorcnt(0);
    }
  }
  __syncthreads();
  if (e >= n) return;
  if (full) {
    // Explicit LDS read-back: the compiler never sees a store to xs (the TDM
    // engine wrote it), so a plain C++ load is legally foldable to undef.
    // Inline asm with a memory clobber keeps it real and ordered.
    unsigned off = (unsigned)(size_t)(const void*)xs + ((unsigned)tid << 4);
    asm volatile("ds_load_b128 %0, %1\n\t"
                 "s_wait_dscnt 0x0"
                 : "=v"(xv)
                 : "v"(off)
                 : "memory");
  } else {
    xv = __builtin_nontemporal_load((const v4f*)x + e);
  }
#else
  if (e >= n) return;
  xv = __builtin_nontemporal_load((const v4f*)x + e);
#endif

  const float x0 = xv.x, x1 = xv.y, x2v = xv.z, x3 = xv.w;

  const float cx0 = cos01(x0);
  const float cx1 = cos01(x1);
  const float cx2 = cos01(x2v);
  const float cx3 = cos01(x3);

  const float pc0 = 0.5f * (1.0f + cx0);
  const float ps0 = 0.5f * (1.0f - cx0);
  const float pc2 = 0.5f * (1.0f + cx2);
  const float ps2 = 0.5f * (1.0f - cx2);
  const float pc3 = 0.5f * (1.0f + cx3);
  const float ps3 = 0.5f * (1.0f - cx3);

  const float G[4]  = {pc2 * pc3, pc2 * ps3, ps2 * pc3, ps2 * ps3};
  const float xc[4] = {x0, x1, x2v, x3};

  // w[d-1] = sum_j G_j * (2*T_d(x_j)^2 - 1)
  float w[4] = {0.0f, 0.0f, 0.0f, 0.0f};
#pragma unroll
  for (int j = 0; j < 4; ++j) {
    const float xj  = xc[j];
    const float tx2 = xj + xj;
    const float t1  = xj;
    const float t2  = fmaf(tx2, t1, -1.0f);   // T2 = 2x^2 - 1
    const float t3  = fmaf(tx2, t2, -t1);     // T3 = 2x*T2 - T1
    const float t4  = fmaf(tx2, t3, -t2);     // T4 = 2x*T3 - T2
    const float g   = G[j];
    w[0] = fmaf(g, fmaf(t1 + t1, t1, -1.0f), w[0]);
    w[1] = fmaf(g, fmaf(t2 + t2, t2, -1.0f), w[1]);
    w[2] = fmaf(g, fmaf(t3 + t3, t3, -1.0f), w[2]);
    w[3] = fmaf(g, fmaf(t4 + t4, t4, -1.0f), w[3]);
  }

  v4f* ov = (v4f*)out + (size_t)e * 4;
#pragma unroll
  for (int d = 0; d < 4; ++d) {
    const float z1 = cx1 * fmaf(pc0, w[d], ps0);
    v4f o = {cx0, z1, cx2, cx3};
    __builtin_nontemporal_store(o, ov + d);  // global_store_b128 th:TH_NT
  }
}

extern "C" void kernel_launch(void* const* d_in, const int* in_sizes, int n_in,
                              void* d_out, int out_size, void* d_ws, size_t ws_size,
                              hipStream_t stream) {
  const float* x = (const float*)d_in[0];
  float* out = (float*)d_out;
  const int n = in_sizes[0] / 4;  // batch elements (x is [B,4] flat)
  const int blocks = (n + 255) / 256;
  qsvt_cheb_kernel<<<blocks, 256, 0, stream>>>(x, out, n);
}